// SheafGluingPoly_42906723287396
// MI455X (gfx1250) — compile-verified
//
#include <hip/hip_runtime.h>
#include <stdint.h>

typedef __attribute__((ext_vector_type(2))) float v2f;
typedef __attribute__((ext_vector_type(8))) float v8f;

// D = A(16x4) * B(4x16) + C, f32, wave32.
#define WMMA4(A, B, C) \
  __builtin_amdgcn_wmma_f32_16x16x4_f32(false, (A), false, (B), (short)0, (C), false, false)

namespace {
constexpr int kB = 4;
constexpr int kM = 50000;
constexpr int kD = 8;
constexpr int kE = 1600000;
constexpr int kNP = kE / 2;        // edge pairs
constexpr int kN = kB * kM * kD;   // 1,600,000 state elements
constexpr float kLam = 1.0f;
}

// out = pc[0] * c0 ; va = 0
__global__ __launch_bounds__(256) void sheaf_init(const float* __restrict__ c0,
                                                  const float* __restrict__ pc,
                                                  float* __restrict__ out,
                                                  float* __restrict__ va) {
  float p0 = pc[0];
  for (int i = blockIdx.x * blockDim.x + threadIdx.x; i < kN; i += gridDim.x * blockDim.x) {
    out[i] = p0 * c0[i];
    va[i] = 0.0f;
  }
}

// v = LAM * v (in place, becomes v_k) ; out += pc[k] * v ; optionally zero the other buffer
__global__ __launch_bounds__(256) void sheaf_axpy(const float* __restrict__ pc, int k,
                                                  float* __restrict__ out,
                                                  float* __restrict__ v,
                                                  float* __restrict__ zbuf) {
  float c = pc[k];
  for (int i = blockIdx.x * blockDim.x + threadIdx.x; i < kN; i += gridDim.x * blockDim.x) {
    float vk = kLam * v[i];
    v[i] = vk;
    out[i] += c * vk;
    if (zbuf) zbuf[i] = 0.0f;
  }
}

// One Laplacian edge pass: acc[:,src,:] += R_src^T r ; acc[:,dst,:] += -R_dst^T r
// with r = R_src p_src - R_dst p_dst.  2 edges per wave iteration, WMMA f32 16x16x4.
__global__ __launch_bounds__(256) void sheaf_edge(const float* __restrict__ p,
                                                  const int* __restrict__ src,
                                                  const int* __restrict__ dst,
                                                  const float* __restrict__ Rs,
                                                  const float* __restrict__ Rd,
                                                  float* __restrict__ acc) {
  // Per-wave double buffer: [0..127] = R_src{e0,e1}, [128..255] = R_dst{e0,e1}
  __shared__ float lds[8][2][256];
  const int lane = threadIdx.x & 31;
  const int w = threadIdx.x >> 5;
  const int wave = blockIdx.x * (blockDim.x >> 5) + w;
  const int nwaves = gridDim.x * (blockDim.x >> 5);

  const int half = lane >> 4;      // K-halves of WMMA fragments
  const int nn = lane & 15;        // N / M-within-16
  const int row = lane & 7;        // R row for stage-1 A
  const int esel = (lane >> 3) & 1;   // stage-1 A edge
  const int b = lane & 3;             // batch
  const int en = (lane >> 2) & 1;     // stage-1 B edge (cols 8..15 mirror 0..7)
  const int esel2 = (nn >> 2) & 1;    // stage-2 output edge
  const bool outact = nn < 8;

  auto issue = [&](int pe, int buf) {
    // 512B per matrix pair, 16B per lane, one async b128 per matrix.
    const float* gs = Rs + (size_t)pe * 128 + lane * 4;
    const float* gd = Rd + (size_t)pe * 128 + lane * 4;
    uint32_t ls = (uint32_t)(uintptr_t)&lds[w][buf][lane * 4];
    uint32_t ld = (uint32_t)(uintptr_t)&lds[w][buf][128 + lane * 4];
    asm volatile("global_load_async_to_lds_b128 %0, %1, off"
                 :: "v"(ls), "v"((uint64_t)(uintptr_t)gs) : "memory");
    asm volatile("global_load_async_to_lds_b128 %0, %1, off"
                 :: "v"(ld), "v"((uint64_t)(uintptr_t)gd) : "memory");
  };

  if (wave < kNP) issue(wave, 0);
  int bufc = 0;
  for (int pe = wave; pe < kNP; pe += nwaves) {
    int pn = pe + nwaves;
    if (pn >= kNP) pn = pe;  // harmless re-load on last trip
    // ensure prior iteration's ds reads of the buffer we are about to overwrite are done
    asm volatile("s_wait_dscnt 0" ::: "memory");
    issue(pn, bufc ^ 1);

    const int e0 = 2 * pe;
    const int s0 = src[e0], s1 = src[e0 + 1];
    const int d0 = dst[e0], d1 = dst[e0 + 1];

    // ---- stage-1 B: gathered p, B[k,n], K rows 0..7 = p_src, 8..15 = p_dst ----
    const int nodeS = en ? s1 : s0;
    const int nodeD = en ? d1 : d0;
    const float* ps = p + ((size_t)b * kM + nodeS) * kD;
    const float* pd = p + ((size_t)b * kM + nodeD) * kD;
    v2f b1_0 = *(const v2f*)(ps + half * 2);
    v2f b1_1 = *(const v2f*)(ps + 4 + half * 2);
    v2f b1_2 = *(const v2f*)(pd + half * 2);
    v2f b1_3 = *(const v2f*)(pd + 4 + half * 2);

    // current buffer's async copy complete (next's 2 loads may stay outstanding)
    asm volatile("s_wait_asynccnt 2" ::: "memory");

    // ---- stage-1 A: rows of [R_src | -R_dst], edge by M-half ----
    const float* lb = &lds[w][bufc][0];
    const float* rs_row = lb + esel * 64 + row * 8;
    const float* rd_row = lb + 128 + esel * 64 + row * 8;
    v2f a1_0 = *(const v2f*)(rs_row + half * 2);
    v2f a1_1 = *(const v2f*)(rs_row + 4 + half * 2);
    v2f a1_2 = -*(const v2f*)(rd_row + half * 2);
    v2f a1_3 = -*(const v2f*)(rd_row + 4 + half * 2);

    v8f r = {};
    r = WMMA4(a1_0, b1_0, r);
    r = WMMA4(a1_1, b1_1, r);
    r = WMMA4(a1_2, b1_2, r);
    r = WMMA4(a1_3, b1_3, r);
    // r_e0[a,b] = reg a @ lane b ; r_e1[a,b] = reg a @ lane 20+b

    // ---- stage-2 B: r_comb cols 0..3 = r_e0, 4..7 = r_e1 (via bpermute) ----
    const int srcl = (nn < 4) ? nn : nn + 16;
    float lo0 = __shfl(r[0], srcl, 32), hi0 = __shfl(r[2], srcl, 32);
    float lo1 = __shfl(r[1], srcl, 32), hi1 = __shfl(r[3], srcl, 32);
    float lo2 = __shfl(r[4], srcl, 32), hi2 = __shfl(r[6], srcl, 32);
    float lo3 = __shfl(r[5], srcl, 32), hi3 = __shfl(r[7], srcl, 32);
    v2f b2_0 = { half ? hi0 : lo0, half ? hi1 : lo1 };
    v2f b2_1 = { half ? hi2 : lo2, half ? hi3 : lo3 };

    // ---- stage-2 A: [R_src^T ; -R_dst^T] per edge, column reads from LDS ----
    const float* tb = (nn < 8) ? lb : (lb + 128);
    const int mm = nn & 7;
    const float sgn = (nn < 8) ? 1.0f : -1.0f;
    const int k0 = half * 2;
    const int k1 = 4 + half * 2;
    v2f aA0 = { sgn * tb[k0 * 8 + mm],        sgn * tb[(k0 + 1) * 8 + mm] };
    v2f aA1 = { sgn * tb[k1 * 8 + mm],        sgn * tb[(k1 + 1) * 8 + mm] };
    v2f aB0 = { sgn * tb[64 + k0 * 8 + mm],   sgn * tb[64 + (k0 + 1) * 8 + mm] };
    v2f aB1 = { sgn * tb[64 + k1 * 8 + mm],   sgn * tb[64 + (k1 + 1) * 8 + mm] };

    v8f cA = {};
    cA = WMMA4(aA0, b2_0, cA);
    cA = WMMA4(aA1, b2_1, cA);
    v8f cB = {};
    cB = WMMA4(aB0, b2_0, cB);
    cB = WMMA4(aB1, b2_1, cB);

    // ---- scatter: lanes 0-7 -> c_src (edge by bit2), lanes 16-23 -> c_dst ----
    const int nodeOut = half ? (esel2 ? d1 : d0) : (esel2 ? s1 : s0);
    float* ap = acc + ((size_t)b * kM + nodeOut) * kD;
    if (outact) {
#pragma unroll
      for (int j = 0; j < 8; ++j) {
        float val = esel2 ? cB[j] : cA[j];
        unsafeAtomicAdd(ap + j, val);
      }
    }
    bufc ^= 1;
  }
}

extern "C" void kernel_launch(void* const* d_in, const int* in_sizes, int n_in,
                              void* d_out, int out_size, void* d_ws, size_t ws_size,
                              hipStream_t stream) {
  const float* c0 = (const float*)d_in[0];
  const int* src = (const int*)d_in[1];
  const int* dst = (const int*)d_in[2];
  const float* Rs = (const float*)d_in[3];
  const float* Rd = (const float*)d_in[4];
  const float* pc = (const float*)d_in[5];
  float* out = (float*)d_out;
  float* va = (float*)d_ws;
  float* vb = va + kN;   // needs 2 * 6.4 MB of workspace

  const int ew_blocks = 6250;   // 6250*256 == kN exactly
  const int edge_blocks = 1024; // 8192 waves, ~98 edge-pairs each

  sheaf_init<<<ew_blocks, 256, 0, stream>>>(c0, pc, out, va);

  sheaf_edge<<<edge_blocks, 256, 0, stream>>>(c0, src, dst, Rs, Rd, va);
  sheaf_axpy<<<ew_blocks, 256, 0, stream>>>(pc, 1, out, va, vb);

  sheaf_edge<<<edge_blocks, 256, 0, stream>>>(va, src, dst, Rs, Rd, vb);
  sheaf_axpy<<<ew_blocks, 256, 0, stream>>>(pc, 2, out, vb, va);

  sheaf_edge<<<edge_blocks, 256, 0, stream>>>(vb, src, dst, Rs, Rd, va);
  sheaf_axpy<<<ew_blocks, 256, 0, stream>>>(pc, 3, out, va, nullptr);
}